// ElementalGTOLogNormal_86723979641149
// MI455X (gfx1250) — compile-verified
//
#include <hip/hip_runtime.h>

typedef __attribute__((ext_vector_type(2))) float v2f;
typedef __attribute__((ext_vector_type(8))) float v8f;

#define NATOM 256
#define NBATCH 16
#define NG_ 20
#define FPSZ 600

__device__ __forceinline__ int species_of(int z) {
    // SPECIES = [1, 6, 7, 8] -> 0..3
    return (z == 1) ? 0 : ((z == 6) ? 1 : ((z == 7) ? 2 : 3));
}

// Reduce T_s[a,g] (held in 4 v8f accumulators, WMMA D layout) into the 30
// fingerprint values for this lane's g column and store them.
//   D layout: VGPR r = T[a=r, g=lane] (lanes 0-15), T[a=r+8, g=lane-16] (lanes 16-31)
// Angular table (a: l, w): a0:(0,1) a1-3:(1,1) a4:(2,1) a5:(2,2) a6:(2,1)
//                          a7:(2,2) a8:(2,2) a9:(2,1); a>=10 unused (T==0).
__device__ __forceinline__ void reduce_and_store(
    const v8f* t, bool hi, int g, bool gv, float* __restrict__ outBase)
{
    float P0[10], P1[10], P2[10];
#pragma unroll
    for (int c = 0; c < 10; ++c) { P0[c] = 0.f; P1[c] = 0.f; P2[c] = 0.f; }
    const float lo = hi ? 0.f : 1.f;
    const float hm = hi ? 1.f : 0.f;
#pragma unroll
    for (int r = 0; r < 8; ++r) {
        const float t0 = t[0][r], t1 = t[1][r], t2 = t[2][r], t3 = t[3][r];
        float v[10];
        v[0] = t0 * t0; v[1] = t1 * t1; v[2] = t2 * t2; v[3] = t3 * t3;
        v[4] = 2.f * t0 * t1; v[5] = 2.f * t0 * t2; v[6] = 2.f * t0 * t3;
        v[7] = 2.f * t1 * t2; v[8] = 2.f * t1 * t3; v[9] = 2.f * t2 * t3;
        if (r == 0) {            // lo: a=0 (l0,w1)   hi: a=8 (l2,w2)
#pragma unroll
            for (int c = 0; c < 10; ++c) { P0[c] += lo * v[c]; P2[c] += hm * 2.f * v[c]; }
        } else if (r == 1) {     // lo: a=1 (l1,w1)   hi: a=9 (l2,w1)
#pragma unroll
            for (int c = 0; c < 10; ++c) { P1[c] += lo * v[c]; P2[c] += hm * v[c]; }
        } else if (r <= 3) {     // lo: a=2,3 (l1,w1) hi: a>=10 -> T==0
#pragma unroll
            for (int c = 0; c < 10; ++c) { P1[c] += lo * v[c]; }
        } else {                 // lo: a=4..7 (l2, w={1,2,1,2})
            const float w = (r == 5 || r == 7) ? 2.f : 1.f;
#pragma unroll
            for (int c = 0; c < 10; ++c) { P2[c] += lo * w * v[c]; }
        }
    }
    // combine the two half-waves (a=0..7 with a=8..15 partials for same g)
#pragma unroll
    for (int c = 0; c < 10; ++c) {
        P0[c] += __shfl_xor(P0[c], 16, 32);
        P1[c] += __shfl_xor(P1[c], 16, 32);
        P2[c] += __shfl_xor(P2[c], 16, 32);
    }
    if (!hi && gv) {
#pragma unroll
        for (int c = 0; c < 10; ++c) {
            outBase[        c * 20 + g] = P0[c];   // LWEIGHTS == 1
            outBase[200 +   c * 20 + g] = P1[c];
            outBase[400 +   c * 20 + g] = P2[c];
        }
    }
}

__global__ __launch_bounds__(128) void gto_fp_kernel(
    const float* __restrict__ coords,   // (B, N, 3)
    const int*   __restrict__ charges,  // (B, N)
    const int*   __restrict__ natoms,   // (B,)
    float*       __restrict__ out)      // (B, N, 600)
{
    __shared__ float sx[NATOM], sy[NATOM], sz[NATOM];
    __shared__ int   ssp[NATOM];

    const int lane = threadIdx.x & 31;
    const bool hi  = lane >= 16;
    const int wave = threadIdx.x >> 5;
    const int atom = blockIdx.x * 4 + wave;   // 4 waves per block, 1 atom per wave
    const int b = atom >> 8;                  // 64 blocks per batch -> whole block same b
    const int i = atom & (NATOM - 1);

    for (int t = threadIdx.x; t < NATOM; t += 128) {
        const float* cp = coords + (size_t)(b * NATOM + t) * 3;
        sx[t] = cp[0]; sy[t] = cp[1]; sz[t] = cp[2];
        ssp[t] = species_of(charges[b * NATOM + t]);
    }
    __syncthreads();

    const int na = natoms[b];
    const bool iValid = (i < na);
    const float xi = sx[i], yi = sy[i], zi = sz[i];

    // per-lane angular component (a = lane & 15), exponents/ids bit-packed
    const int a  = lane & 15;
    const int nx = (0x4604  >> (2 * a)) & 3;   // x exponent
    const int my = (0x12410 >> (2 * a)) & 3;   // y exponent
    const int kz = (0x94040 >> (2 * a)) & 3;   // z exponent
    const int li = (0xAAA54 >> (2 * a)) & 3;   // l index
    const bool aValid = (a < 10);

    // per-lane Gaussian columns: tile0 -> g0 = lane&15, tile1 -> g0+16
    const int g0 = lane & 15;
    const int g1 = g0 + 16;
    const bool g1v = (g1 < NG_);
    const float off0 = 0.3f * (float)(g0 + 1);
    const float off1 = 0.3f * (float)((g1v ? g1 : (NG_ - 1)) + 1);
    const float lg0 = __logf(off0), io0 = 1.0f / off0;
    const float lg1 = __logf(off1), io1 = 1.0f / off1;

    const v8f vz = {0.f, 0.f, 0.f, 0.f, 0.f, 0.f, 0.f, 0.f};
    v8f acc0[4] = {vz, vz, vz, vz};   // T_s, g = 0..15
    v8f acc1[4] = {vz, vz, vz, vz};   // T_s, g = 16..19

    for (int ch = 0; ch < NATOM / 4; ++ch) {
        // A(16x4) K-mapping: lanes 0-15 hold K=0,1 ; lanes 16-31 hold K=2,3
        const int jb = ch * 4 + (hi ? 2 : 0);
        float ang[2], rg0[2], rg1[2];
        int sj[2];
#pragma unroll
        for (int jj = 0; jj < 2; ++jj) {
            const int j = jb + jj;
            const float dx = xi - sx[j], dy = yi - sy[j], dz = zi - sz[j];
            const float d2 = dx * dx + dy * dy + dz * dz;
            const bool ok = iValid && (j < na) && (j != i) && (d2 < 36.0f);
            const float d2s = ok ? d2 : 1.0f;                 // avoid NaN when masked
            const float d = __fsqrt_rn(d2s);
            const float sigma2 = __logf(1.0f + 2.0f / d2s);   // log1p(W/d^2)
            const float mu = __logf(d) - 0.5f * sigma2;       // log(d/sqrt(1+W/d^2))
            const float cut = 0.5f * (__cosf(d * 0.52359878f) + 1.0f);  // pi/HIGH
            const float pref = ok ? (cut * __frsqrt_rn(sigma2) * 0.56418958f) : 0.0f;
            const float inv2s = 0.5f / sigma2;
            const float c0 = lg0 - mu, c1 = lg1 - mu;
            rg0[jj] = pref * io0 * __expf(-c0 * c0 * inv2s);
            rg1[jj] = g1v ? (pref * io1 * __expf(-c1 * c1 * inv2s)) : 0.0f;
            const float invd = ok ? (1.0f / d) : 0.0f;        // carries the mask
            const float id2 = invd * invd;
            const float scale = id2 * ((li == 0) ? 1.0f : ((li == 1) ? invd : id2));
            const float px = (nx == 0) ? 1.0f : ((nx == 1) ? dx : dx * dx);
            const float py = (my == 0) ? 1.0f : ((my == 1) ? dy : dy * dy);
            const float pz = (kz == 0) ? 1.0f : ((kz == 1) ? dz : dz * dz);
            ang[jj] = aValid ? (scale * px * py * pz) : 0.0f; // zero rows a>=10
            sj[jj] = ssp[j];
        }
        v2f A; A.x = ang[0]; A.y = ang[1];
#pragma unroll
        for (int s = 0; s < 4; ++s) {
            v2f B0, B1;   // B(4x16): row K = neighbor, col N = Gaussian
            B0.x = (sj[0] == s) ? rg0[0] : 0.0f;
            B0.y = (sj[1] == s) ? rg0[1] : 0.0f;
            B1.x = (sj[0] == s) ? rg1[0] : 0.0f;
            B1.y = (sj[1] == s) ? rg1[1] : 0.0f;
            acc0[s] = __builtin_amdgcn_wmma_f32_16x16x4_f32(
                false, A, false, B0, (short)0, acc0[s], false, false);
            acc1[s] = __builtin_amdgcn_wmma_f32_16x16x4_f32(
                false, A, false, B1, (short)0, acc1[s], false, false);
        }
    }

    float* outBase = out + (size_t)atom * FPSZ;
    reduce_and_store(acc0, hi, g0, true, outBase);
    reduce_and_store(acc1, hi, g1, g1v, outBase);
}

extern "C" void kernel_launch(void* const* d_in, const int* in_sizes, int n_in,
                              void* d_out, int out_size, void* d_ws, size_t ws_size,
                              hipStream_t stream) {
    const float* coords  = (const float*)d_in[0];   // (16,256,3) f32
    const int*   charges = (const int*)d_in[1];     // (16,256) i32
    const int*   natoms  = (const int*)d_in[2];     // (16,) i32
    float* out = (float*)d_out;                     // (16,256,600) f32

    dim3 grid((NBATCH * NATOM) / 4);   // 1024 blocks, 4 atoms (waves) each
    dim3 block(128);
    hipLaunchKernelGGL(gto_fp_kernel, grid, block, 0, stream,
                       coords, charges, natoms, out);
}